// DistanceEstimator_65060164600407
// MI455X (gfx1250) — compile-verified
//
#include <hip/hip_runtime.h>

#define N_NODES 100000
#define N_EDGES 3200000
#define FDIM 128
#define H 64
#define BATCH 128

typedef __attribute__((ext_vector_type(2))) float v2f;
typedef __attribute__((ext_vector_type(8))) float v8f;

// ---------------------------------------------------------------- utilities
__global__ void zero_kernel(float* __restrict__ p, long long n) {
    long long i = (long long)blockIdx.x * blockDim.x + threadIdx.x;
    if (i < n) p[i] = 0.0f;
}

// ------------------------------------------------- edge scatter, layer 1 (F=128)
// 32 threads per edge, 4 contiguous floats per thread. agg[dst] += x[src]; cnt[dst] += 1.
__global__ void scatter_f128(const float* __restrict__ x, const int* __restrict__ ei,
                             float* __restrict__ agg, float* __restrict__ cnt) {
    long long t = (long long)blockIdx.x * blockDim.x + threadIdx.x;
    long long e = t >> 5;
    if (e >= N_EDGES) return;
    int lane = (int)(t & 31);
    int src = ei[e];
    int dst = ei[N_EDGES + e];
    const float* xs = x + (long long)src * FDIM + lane * 4;
    float* ad = agg + (long long)dst * FDIM + lane * 4;
#pragma unroll
    for (int k = 0; k < 4; ++k) unsafeAtomicAdd(ad + k, xs[k]);
    if (lane == 0) unsafeAtomicAdd(cnt + dst, 1.0f);
}

// ------------------------------------------------- edge scatter, layer 2 (F=64)
__global__ void scatter_f64(const float* __restrict__ h, const int* __restrict__ ei,
                            float* __restrict__ agg) {
    long long t = (long long)blockIdx.x * blockDim.x + threadIdx.x;
    long long e = t >> 4;
    if (e >= N_EDGES) return;
    int lane = (int)(t & 15);
    int src = ei[e];
    int dst = ei[N_EDGES + e];
    const float* xs = h + (long long)src * H + lane * 4;
    float* ad = agg + (long long)dst * H + lane * 4;
#pragma unroll
    for (int k = 0; k < 4; ++k) unsafeAtomicAdd(ad + k, xs[k]);
}

// ------------------------------------------------- SAGE layer GEMM via fp32 WMMA
// out[i,:] = relu( (agg[i,:]/max(cnt[i],1)) @ Wl + bl + xin[i,:] @ Wr ),  out is [N, 64].
// Block = 128 threads = 4 waves; wave w handles column tile [16w, 16w+16); block row strip = 16 nodes.
// V_WMMA_F32_16X16X4_F32 fragment layouts per ISA 7.12.2:
//   A (16x4):  row = lane&15 (both halves); VGPR0 holds K = k0+2*half, VGPR1 holds K = k0+2*half+1
//   B (4x16):  col = lane&15; same K striping as A (row-of-B striped across lanes per VGPR)
//   C (16x16): VGPR r, lane L -> row = r + 8*(L>=16), col = L&15
template <int F>
__global__ void sage_gemm(const float* __restrict__ agg, const float* __restrict__ cnt,
                          const float* __restrict__ xin,
                          const float* __restrict__ Wl, const float* __restrict__ bl,
                          const float* __restrict__ Wr, float* __restrict__ out) {
    int wave = threadIdx.x >> 5;      // 0..3 : column tile
    int lane = threadIdx.x & 31;
    int half = lane >> 4;             // 0 or 1
    int l15  = lane & 15;
    int row  = blockIdx.x * 16 + l15; // A-matrix row for this lane
    int col  = wave * 16 + l15;       // B/C column for this lane

    float ic = 1.0f / fmaxf(cnt[row], 1.0f);
    const float* arow = agg + (long long)row * F;
    const float* xrow = xin + (long long)row * F;

    v8f c = {};
#pragma unroll
    for (int k0 = 0; k0 < F; k0 += 4) {
        int ka = k0 + half * 2;
        v2f a, b;
        a.x = arow[ka] * ic;
        a.y = arow[ka + 1] * ic;
        b.x = Wl[ka * H + col];
        b.y = Wl[(ka + 1) * H + col];
        c = __builtin_amdgcn_wmma_f32_16x16x4_f32(false, a, false, b, (short)0, c, false, false);
    }
#pragma unroll
    for (int k0 = 0; k0 < F; k0 += 4) {
        int ka = k0 + half * 2;
        v2f a, b;
        a.x = xrow[ka];
        a.y = xrow[ka + 1];
        b.x = Wr[ka * H + col];
        b.y = Wr[(ka + 1) * H + col];
        c = __builtin_amdgcn_wmma_f32_16x16x4_f32(false, a, false, b, (short)0, c, false, false);
    }

    float bias = bl[col];
#pragma unroll
    for (int r = 0; r < 8; ++r) {
        int orow = blockIdx.x * 16 + r + half * 8;
        out[(long long)orow * H + col] = fmaxf(c[r] + bias, 0.0f);
    }
}

// ------------------------------------------------- global mean pool (scatter)
__global__ void pool_scatter(const float* __restrict__ h, const int* __restrict__ batch,
                             float* __restrict__ pool, float* __restrict__ pcnt) {
    long long t = (long long)blockIdx.x * blockDim.x + threadIdx.x;
    if (t >= (long long)N_NODES * H) return;
    int i = (int)(t >> 6);
    int f = (int)(t & 63);
    int b = batch[i];
    unsafeAtomicAdd(pool + b * H + f, h[t]);
    if (f == 0) unsafeAtomicAdd(pcnt + b, 1.0f);
}

__global__ void pool_div(const float* __restrict__ pool, const float* __restrict__ pcnt,
                         float* __restrict__ emb) {
    int t = blockIdx.x * blockDim.x + threadIdx.x;
    if (t >= BATCH * H) return;
    int b = t >> 6;
    emb[t] = pool[t] / fmaxf(pcnt[b], 1.0f);
}

// ------------------------------------------------- depth normalization (B=128)
__global__ void depth_norm(const float* __restrict__ depth, float* __restrict__ dn) {
    __shared__ float s1[BATCH], s2[BATCH];
    int t = threadIdx.x;
    float v = depth[t];
    s1[t] = v;
    s2[t] = v * v;
    __syncthreads();
    for (int o = BATCH / 2; o > 0; o >>= 1) {
        if (t < o) { s1[t] += s1[t + o]; s2[t] += s2[t + o]; }
        __syncthreads();
    }
    float mean = s1[0] / (float)BATCH;
    float var  = s2[0] / (float)BATCH - mean * mean;
    float sd   = sqrtf(fmaxf(var, 0.0f));
    dn[t] = (v - mean) / (sd + 1e-6f);
}

// ------------------------------------------------- final MLP: [B,129]->64 relu ->1
__global__ void mlp_kernel(const float* __restrict__ se, const float* __restrict__ ge,
                           const float* __restrict__ dn,
                           const float* __restrict__ W1, const float* __restrict__ b1,
                           const float* __restrict__ W2, const float* __restrict__ b2,
                           float* __restrict__ out) {
    int b = blockIdx.x;   // 0..127
    int j = threadIdx.x;  // 0..63
    float acc = b1[j];
#pragma unroll
    for (int k = 0; k < H; ++k) acc += se[b * H + k] * W1[k * H + j];
#pragma unroll
    for (int k = 0; k < H; ++k) acc += ge[b * H + k] * W1[(H + k) * H + j];
    acc += dn[b] * W1[(2 * H) * H + j];
    acc = fmaxf(acc, 0.0f);
    __shared__ float sh[H];
    sh[j] = acc * W2[j];
    __syncthreads();
    for (int o = H / 2; o > 0; o >>= 1) {
        if (j < o) sh[j] += sh[j + o];
        __syncthreads();
    }
    if (j == 0) out[b] = sh[0] + b2[0];
}

// ---------------------------------------------------------------- launcher
extern "C" void kernel_launch(void* const* d_in, const int* in_sizes, int n_in,
                              void* d_out, int out_size, void* d_ws, size_t ws_size,
                              hipStream_t stream) {
    (void)in_sizes; (void)n_in; (void)out_size; (void)ws_size;

    // workspace layout (floats)
    float* ws   = (float*)d_ws;
    float* agg1 = ws;                                      // N*128  (reused: agg2 = agg1, h2 = agg1 + N*64)
    float* h1   = agg1 + (size_t)N_NODES * FDIM;           // N*64
    float* cnt  = h1 + (size_t)N_NODES * H;                // N
    float* pool = cnt + N_NODES;                           // B*64
    float* pcnt = pool + BATCH * H;                        // B
    float* semb = pcnt + BATCH;                            // B*64
    float* gemb = semb + BATCH * H;                        // B*64
    float* dn   = gemb + BATCH * H;                        // B
    float* agg2 = agg1;
    float* h2   = agg1 + (size_t)N_NODES * H;

    const int ZT = 256;
    for (int g = 0; g < 2; ++g) {
        const float* x   = (const float*)d_in[g ? 3 : 0];
        const int*   ei  = (const int*)  d_in[g ? 4 : 1];
        const int*   bt  = (const int*)  d_in[g ? 5 : 2];
        const float* Wl1 = (const float*)d_in[g ? 13 : 7];
        const float* bl1 = (const float*)d_in[g ? 14 : 8];
        const float* Wr1 = (const float*)d_in[g ? 15 : 9];
        const float* Wl2 = (const float*)d_in[g ? 16 : 10];
        const float* bl2 = (const float*)d_in[g ? 17 : 11];
        const float* Wr2 = (const float*)d_in[g ? 18 : 12];
        float* emb = g ? gemb : semb;

        // layer 1: mean aggregate over 128 feats + degree counts
        long long nz1 = (long long)N_NODES * FDIM;
        zero_kernel<<<(unsigned)((nz1 + ZT - 1) / ZT), ZT, 0, stream>>>(agg1, nz1);
        zero_kernel<<<(N_NODES + ZT - 1) / ZT, ZT, 0, stream>>>(cnt, N_NODES);
        long long t1 = (long long)N_EDGES * 32;
        scatter_f128<<<(unsigned)((t1 + ZT - 1) / ZT), ZT, 0, stream>>>(x, ei, agg1, cnt);
        sage_gemm<FDIM><<<N_NODES / 16, 128, 0, stream>>>(agg1, cnt, x, Wl1, bl1, Wr1, h1);

        // layer 2: same counts, 64-feat aggregate
        long long nz2 = (long long)N_NODES * H;
        zero_kernel<<<(unsigned)((nz2 + ZT - 1) / ZT), ZT, 0, stream>>>(agg2, nz2);
        long long t2 = (long long)N_EDGES * 16;
        scatter_f64<<<(unsigned)((t2 + ZT - 1) / ZT), ZT, 0, stream>>>(h1, ei, agg2);
        sage_gemm<H><<<N_NODES / 16, 128, 0, stream>>>(agg2, cnt, h1, Wl2, bl2, Wr2, h2);

        // global mean pool
        zero_kernel<<<(BATCH * H + BATCH + ZT - 1) / ZT, ZT, 0, stream>>>(pool, BATCH * H + BATCH);
        long long tp = (long long)N_NODES * H;
        pool_scatter<<<(unsigned)((tp + ZT - 1) / ZT), ZT, 0, stream>>>(h2, bt, pool, pcnt);
        pool_div<<<(BATCH * H + ZT - 1) / ZT, ZT, 0, stream>>>(pool, pcnt, emb);
    }

    // depth normalization + MLP head
    const float* depth = (const float*)d_in[6];
    const float* W1 = (const float*)d_in[19];
    const float* b1 = (const float*)d_in[20];
    const float* W2 = (const float*)d_in[21];
    const float* b2 = (const float*)d_in[22];
    depth_norm<<<1, BATCH, 0, stream>>>(depth, dn);
    mlp_kernel<<<BATCH, H, 0, stream>>>(semb, gemb, dn, W1, b1, W2, b2, (float*)d_out);
}